// RetentionPolicy_9311489098312
// MI455X (gfx1250) — compile-verified
//
#include <hip/hip_runtime.h>
#include <stdint.h>

// RetentionPolicy (training branch) for MI455X / gfx1250.
// Memory-bound selection: 512 MiB streamed once -> ~23 us floor at 23.3 TB/s.
// One workgroup (256 threads, 8 wave32) per row; row staged in LDS via the
// CDNA5 async-to-LDS path (ASYNCcnt, non-temporal), selection via LDS
// radix-select + block scans. No reuse -> th:TH_LOAD_NT to spare L2.

#define BS_ROWS   16384
#define K_TOTAL   8192
#define MAXC      100
#define MINC      20
#define THREADS   256
#define SEG       (K_TOTAL / THREADS)             // 32 contiguous elems/thread
#define CHUNKS16  (K_TOTAL * 4 / 16 / THREADS)    // 8 x b128 async loads/thread

// Monotone float -> uint key: larger float => larger key (handles negatives).
__device__ __forceinline__ unsigned int f2key(float f) {
  unsigned int u = __float_as_uint(f);
  return u ^ ((unsigned int)((int)u >> 31) | 0x80000000u);
}

__global__ __launch_bounds__(THREADS)
void retention_topk_kernel(const float* __restrict__ logits,
                           int* __restrict__ bs_out,
                           int* __restrict__ cls_out,
                           unsigned char* __restrict__ mask_out)
{
  __shared__ float        s_vals[K_TOTAL];   // 32 KB row staging
  __shared__ unsigned int s_hist[256];       // radix histogram
  __shared__ unsigned int s_scan[THREADS];   // block scans
  __shared__ unsigned int s_meta[4];         // 0:prefix 1:k-rem 2:pos-count 3:num_train

  const int row = blockIdx.x;
  const int tid = threadIdx.x;
  const float* rowp = logits + (size_t)row * K_TOTAL;

  // ---- Phase A: stream row into LDS with CDNA5 async-to-LDS b128 copies ----
  {
    unsigned int       lds_base = (unsigned int)(uintptr_t)(&s_vals[0]);
    unsigned long long gbase    = (unsigned long long)(uintptr_t)rowp;
#pragma unroll
    for (int j = 0; j < CHUNKS16; ++j) {
      unsigned int       chunk = (unsigned int)(j * THREADS + tid); // 0..2047
      unsigned int       lds_a = lds_base + chunk * 16u;
      unsigned long long gaddr = gbase + (unsigned long long)chunk * 16ull;
      asm volatile("global_load_async_to_lds_b128 %0, %1, off th:TH_LOAD_NT"
                   :: "v"(lds_a), "v"(gaddr) : "memory");
    }
    asm volatile("s_wait_asynccnt 0" ::: "memory");
  }
  if (tid < 4) s_meta[tid] = 0u;
  s_hist[tid] = 0u;
  __syncthreads();  // all waves' async data visible + hist/meta zeroed

  // ---- Phase B+C3: fused positive-count + radix pass over top byte ----
  {
    unsigned int cpos = 0;
    for (int i = tid; i < K_TOTAL; i += THREADS) {
      float f = s_vals[i];
      cpos += (f >= 0.0f) ? 1u : 0u;                 // prob >= 0.5 <=> logit >= 0
      atomicAdd(&s_hist[f2key(f) >> 24], 1u);
    }
    atomicAdd(&s_meta[2], cpos);
  }
  __syncthreads();
  if (tid == 0) {
    unsigned int nt = s_meta[2];
    nt = nt < MINC ? MINC : (nt > MAXC ? MAXC : nt);
    unsigned int k = nt;
    int b = 255;
    while (b > 0 && s_hist[b] < k) { k -= s_hist[b]; --b; }
    s_meta[0] = (unsigned int)b << 24;
    s_meta[1] = k;          // still needed within chosen bin
    s_meta[3] = nt;
  }
  __syncthreads();
  unsigned int prefix = s_meta[0];

  // ---- Phase C: remaining 3 radix passes for the num_train-th largest ----
#pragma unroll
  for (int p = 2; p >= 0; --p) {
    s_hist[tid] = 0u;
    __syncthreads();
    const int sh = p * 8;
    const unsigned int himask = 0xFFFFFFFFu << (sh + 8);   // p<=2 -> shift<=24
    for (int i = tid; i < K_TOTAL; i += THREADS) {
      unsigned int key = f2key(s_vals[i]);
      if ((key & himask) == prefix)
        atomicAdd(&s_hist[(key >> sh) & 255u], 1u);
    }
    __syncthreads();
    if (tid == 0) {
      unsigned int k = s_meta[1];
      int b = 255;
      while (b > 0 && s_hist[b] < k) { k -= s_hist[b]; --b; }
      s_meta[0] = prefix | ((unsigned int)b << sh);
      s_meta[1] = k;
    }
    __syncthreads();
    prefix = s_meta[0];
  }
  const unsigned int T      = s_meta[0];  // threshold key
  const unsigned int kEq    = s_meta[1];  // #(key==T) to keep (lowest idx first)
  const unsigned int nTrain = s_meta[3];

  // ---- Phase D: stable, index-ascending compaction ----
  const int base = tid * SEG;

  // One LDS sweep: cache this thread's 32 keys in VGPRs.
  unsigned int kseg[SEG];
#pragma unroll
  for (int i = 0; i < SEG; ++i) kseg[i] = f2key(s_vals[base + i]);

  // D1: per-thread equal counts -> scan -> global equal-rank base
  unsigned int eqc = 0;
#pragma unroll
  for (int i = 0; i < SEG; ++i) eqc += (kseg[i] == T) ? 1u : 0u;
  s_scan[tid] = eqc;
  __syncthreads();
  for (int off = 1; off < THREADS; off <<= 1) {
    unsigned int y = (tid >= off) ? s_scan[tid - off] : 0u;
    __syncthreads();
    s_scan[tid] += y;
    __syncthreads();
  }
  const unsigned int eqBase = s_scan[tid] - eqc;
  __syncthreads();

  // D2: per-thread selected counts -> scan -> write positions
  unsigned int selc = 0;
  {
    unsigned int eqr = eqBase;
#pragma unroll
    for (int i = 0; i < SEG; ++i) {
      bool eq  = (kseg[i] == T);
      bool sel = (kseg[i] > T) || (eq && eqr < kEq);
      eqr  += eq  ? 1u : 0u;
      selc += sel ? 1u : 0u;
    }
  }
  s_scan[tid] = selc;
  __syncthreads();
  for (int off = 1; off < THREADS; off <<= 1) {
    unsigned int y = (tid >= off) ? s_scan[tid - off] : 0u;
    __syncthreads();
    s_scan[tid] += y;
    __syncthreads();
  }
  unsigned int wpos = s_scan[tid] - selc;

  // D3: emit selected class indices (ascending by construction)
  const size_t out_row = (size_t)row * MAXC;
  {
    unsigned int eqr = eqBase;
#pragma unroll
    for (int i = 0; i < SEG; ++i) {
      bool eq  = (kseg[i] == T);
      bool sel = (kseg[i] > T) || (eq && eqr < kEq);
      eqr += eq ? 1u : 0u;
      if (sel) cls_out[out_row + (wpos++)] = base + i;
    }
  }

  // ---- Phase E: bs_idx, mask, and -1 padding ----
  for (int j = tid; j < MAXC; j += THREADS) {
    bool v = (unsigned int)j < nTrain;
    bs_out[out_row + j]   = v ? row : -1;
    mask_out[out_row + j] = v ? 1 : 0;
    if (!v) cls_out[out_row + j] = -1;
  }
}

extern "C" void kernel_launch(void* const* d_in, const int* in_sizes, int n_in,
                              void* d_out, int out_size, void* d_ws, size_t ws_size,
                              hipStream_t stream) {
  (void)in_sizes; (void)n_in; (void)out_size; (void)d_ws; (void)ws_size;
  const float* logits = (const float*)d_in[0];

  // Output tuple flattened in return order: bs_idx[i32], cls_idx[i32], mask[i8]
  char* out = (char*)d_out;
  int*           bs   = (int*)out;
  int*           cls  = (int*)(out + (size_t)BS_ROWS * MAXC * sizeof(int));
  unsigned char* mask = (unsigned char*)(out + (size_t)BS_ROWS * MAXC * sizeof(int) * 2);

  hipLaunchKernelGGL(retention_topk_kernel, dim3(BS_ROWS), dim3(THREADS), 0, stream,
                     logits, bs, cls, mask);
}